// KnowledgeRNN_34703335752428
// MI455X (gfx1250) — compile-verified
//
#include <hip/hip_runtime.h>
#include <math.h>

#define SEQ     2048
#define EMB     1024
#define STATE   1024
#define QIN     2048
#define QUERY   256
#define NKB     10000
#define VALUE   512
#define LSTM_IN 1536
#define DEC_IN  2560
#define NTOK    32000

#define NBLK    64          // workgroups cooperating on the recurrent scan

typedef __attribute__((ext_vector_type(2))) float v2f;
typedef __attribute__((ext_vector_type(8))) float v8f;

__device__ __forceinline__ float wave_sum32(float v) {
#pragma unroll
  for (int m = 16; m > 0; m >>= 1) v += __shfl_xor(v, m, 32);
  return v;
}
__device__ __forceinline__ float wave_max32(float v) {
#pragma unroll
  for (int m = 16; m > 0; m >>= 1) v = fmaxf(v, __shfl_xor(v, m, 32));
  return v;
}
__device__ __forceinline__ float sigmoidf(float x) {
  return 1.0f / (1.0f + expf(-x));
}

// Monotonic-counter global barrier for NBLK co-resident blocks.
// Counter is zeroed by k_init at the start of every kernel_launch call.
__device__ __forceinline__ void gbar(unsigned* cnt, unsigned& phase) {
  __threadfence();                    // publish this block's global writes
  __syncthreads();
  if (threadIdx.x == 0) {
    atomicAdd(cnt, 1u);
    phase += NBLK;
    while (__hip_atomic_load(cnt, __ATOMIC_ACQUIRE, __HIP_MEMORY_SCOPE_AGENT) < phase) {
      __builtin_amdgcn_s_sleep(1);
    }
  }
  __syncthreads();
  __threadfence();                    // acquire: discard stale cached lines
}

// ---------------------------------------------------------------------------
// Kernel 0: zero the barrier counter and hx/cx state (deterministic per call).
// ---------------------------------------------------------------------------
__global__ __launch_bounds__(256) void k_init(unsigned* __restrict__ cnt,
                                              float* __restrict__ hx_g,
                                              float* __restrict__ cx_g) {
  if (threadIdx.x == 0) *cnt = 0u;
  for (int i = threadIdx.x; i < STATE; i += 256) {
    hx_g[i] = 0.0f;
    cx_g[i] = 0.0f;
  }
}

// ---------------------------------------------------------------------------
// Kernel 1: embedding gather.  X[s][0:1024] = enc_W[ids[s]]
// ---------------------------------------------------------------------------
__global__ __launch_bounds__(256) void k_embed(const int* __restrict__ ids,
                                               const float* __restrict__ encW,
                                               float* __restrict__ X) {
  int s = blockIdx.x;
  int tok = ids[s];
  const float* src = encW + (size_t)tok * EMB;
  float* dst = X + (size_t)s * DEC_IN;
  for (int c = threadIdx.x; c < EMB; c += 256) dst[c] = src[c];
}

// ---------------------------------------------------------------------------
// Kernel 2: cooperative recurrent scan across NBLK=64 blocks x 1024 threads.
// 4 global barriers per timestep; every weight row is read exactly once per
// step, coalesced, and all weights (~90MB) stay resident in the 192MB L2.
// Stage map:
//  S1: stage hx/x to LDS; qh partials over an 8(jg) x 8(ig) block grid
//  S2: reduce own 32-wide qh chunk (tanh); q partials over own i-chunk
//  S3: reduce q; logits+exp for own KB rows; partial Z and partial val
//  S4: finalize val; own 64 LSTM gate rows {j,j+1K,j+2K,j+3K}; state update
// Softmax is computed without the max-shift (mathematically identical;
// logits here are O(1..10) so exp() is safe in fp32).
// ---------------------------------------------------------------------------
__global__ __launch_bounds__(1024) void k_scan(
    const float* __restrict__ Wq1, const float* __restrict__ bq1,
    const float* __restrict__ Wq2, const float* __restrict__ bq2,
    const float* __restrict__ kbK, const float* __restrict__ kbV,
    const float* __restrict__ Wih, const float* __restrict__ bih,
    const float* __restrict__ Whh, const float* __restrict__ bhh,
    float* __restrict__ X,
    float* __restrict__ qh_part,   // [8][QIN]
    float* __restrict__ q_part,    // [NBLK][QUERY]
    float* __restrict__ vpart_g,   // [NBLK][VALUE]
    float* __restrict__ Zpart_g,   // [NBLK]
    float* __restrict__ hx_g, float* __restrict__ cx_g,
    unsigned* __restrict__ cnt) {
  __shared__ float hx_s[STATE], xv_s[EMB];
  __shared__ float jpart_s[1024];      // reused partial-combine scratch
  __shared__ float qhc_s[32];
  __shared__ float q_s[QUERY];
  __shared__ float e_s[160];
  __shared__ float red_s[32];
  __shared__ float val_s[VALUE];
  __shared__ float gates_s[64];
  __shared__ float zsh;

  const int tid = threadIdx.x;
  const int lane = tid & 31;
  const int wave = tid >> 5;
  const int blk = blockIdx.x;          // 0..63
  unsigned phase = 0;

  const int jg = blk >> 3;             // qh output group (8 x 256)
  const int ig = blk & 7;              // qh input group  (8 x 256)
  const int kmax = (NKB - blk + NBLK - 1) / NBLK;  // KB rows owned (156/157)
  const int j0 = blk * 16;             // owned hx indices [j0, j0+16)

  for (int s = 0; s < SEQ; ++s) {
    float* Xrow = X + (size_t)s * DEC_IN;

    // ------------------- S1: stage state, qh partials -------------------
    hx_s[tid] = hx_g[tid];
    xv_s[tid] = Xrow[tid];
    __syncthreads();
    {
      const int jl = tid & 255;                  // output column in group
      const int isub = tid >> 8;                 // 0..3
      const int i0 = ig * 256 + isub * 64;
      const int j = jg * 256 + jl;
      const float* inp = (i0 < STATE) ? (hx_s + i0) : (xv_s + (i0 - STATE));
      const float* wp = Wq1 + (size_t)i0 * QIN + j;
      float acc = 0.0f;
      for (int t = 0; t < 64; ++t) acc += inp[t] * wp[(size_t)t * QIN];
      jpart_s[isub * 256 + jl] = acc;
    }
    __syncthreads();
    if (tid < 256) {
      qh_part[(size_t)ig * QIN + jg * 256 + tid] =
          jpart_s[tid] + jpart_s[256 + tid] + jpart_s[512 + tid] + jpart_s[768 + tid];
    }
    gbar(cnt, phase);

    // ------------------- S2: qh chunk reduce, q partials ----------------
    if (tid < 32) {
      int j = blk * 32 + tid;
      float acc = bq1[j];
      for (int p = 0; p < 8; ++p) acc += qh_part[(size_t)p * QIN + j];
      qhc_s[tid] = tanhf(acc);
    }
    __syncthreads();
    {
      const int k = tid & 255;
      const int isub = tid >> 8;                 // 0..3, 8 i's each
      float acc = 0.0f;
      const float* wp = Wq2 + (size_t)(blk * 32 + isub * 8) * QUERY + k;
      for (int u = 0; u < 8; ++u) acc += qhc_s[isub * 8 + u] * wp[(size_t)u * QUERY];
      jpart_s[isub * 256 + k] = acc;
    }
    __syncthreads();
    if (tid < 256) {
      q_part[(size_t)blk * QUERY + tid] =
          jpart_s[tid] + jpart_s[256 + tid] + jpart_s[512 + tid] + jpart_s[768 + tid];
    }
    gbar(cnt, phase);

    // --------- S3: q reduce, logits+exp, partial Z, partial val ---------
    if (tid < QUERY) {
      float acc = bq2[tid];
      for (int p = 0; p < NBLK; ++p) acc += q_part[(size_t)p * QUERY + tid];
      q_s[tid] = acc;
    }
    __syncthreads();
    for (int k = wave; k < kmax; k += 32) {
      int r = blk + NBLK * k;
      const float* krow = kbK + (size_t)r * QUERY;
      float p = 0.0f;
#pragma unroll
      for (int c = 0; c < QUERY; c += 32) p += krow[c + lane] * q_s[c + lane];
      p = wave_sum32(p);
      if (lane == 0) e_s[k] = expf(p);
    }
    __syncthreads();
    {
      float lsum = (tid < kmax) ? e_s[tid] : 0.0f;
      lsum = wave_sum32(lsum);
      if (lane == 0) red_s[wave] = lsum;
    }
    __syncthreads();
    if (tid == 0) {
      float z = 0.0f;
      for (int w = 0; w < 32; ++w) z += red_s[w];
      Zpart_g[blk] = z;
    }
    {
      const int c = tid & (VALUE - 1);
      const int h = tid >> 9;                    // 0/1: split rows
      float acc = 0.0f;
      for (int k = h; k < kmax; k += 2) {
        acc += e_s[k] * kbV[(size_t)(blk + NBLK * k) * VALUE + c];
      }
      jpart_s[h * VALUE + c] = acc;
    }
    __syncthreads();
    if (tid < VALUE) {
      vpart_g[(size_t)blk * VALUE + tid] = jpart_s[tid] + jpart_s[VALUE + tid];
    }
    gbar(cnt, phase);

    // ------------- S4: finalize val, gates, state update ----------------
    if (tid == 0) {
      float z = 0.0f;
      for (int p = 0; p < NBLK; ++p) z += Zpart_g[p];
      zsh = 1.0f / z;
    }
    __syncthreads();
    if (tid < VALUE) {
      float acc = 0.0f;
      for (int p = 0; p < NBLK; ++p) acc += vpart_g[(size_t)p * VALUE + tid];
      val_s[tid] = acc * zsh;
    }
    __syncthreads();
    for (int idx = wave; idx < 64; idx += 32) {
      const int qtr = idx >> 4;
      const int t = idx & 15;
      const int row = qtr * STATE + j0 + t;
      const float* wi = Wih + (size_t)row * LSTM_IN;
      const float* wh = Whh + (size_t)row * STATE;
      float acc = 0.0f;
#pragma unroll 4
      for (int i = lane; i < EMB; i += 32)   acc += xv_s[i]  * wi[i];
#pragma unroll 4
      for (int i = lane; i < VALUE; i += 32) acc += val_s[i] * wi[EMB + i];
#pragma unroll 4
      for (int i = lane; i < STATE; i += 32) acc += hx_s[i]  * wh[i];
      acc = wave_sum32(acc);
      if (lane == 0) gates_s[idx] = acc + bih[row] + bhh[row];
    }
    __syncthreads();
    if (tid < 16) {
      const int j = j0 + tid;
      float igt = sigmoidf(gates_s[tid]);
      float fgt = sigmoidf(gates_s[16 + tid]);
      float ggt = tanhf(gates_s[32 + tid]);
      float ogt = sigmoidf(gates_s[48 + tid]);
      float cnew = fgt * cx_g[j] + igt * ggt;
      float hnew = ogt * tanhf(cnew);
      Xrow[EMB + VALUE + j] = hx_s[j];           // PRE-update hidden state
      cx_g[j] = cnew;
      hx_g[j] = hnew;
    }
    if (tid < 8) {
      const int c = blk * 8 + tid;
      Xrow[EMB + c] = val_s[c];                  // kb_out
    }
    gbar(cnt, phase);                            // end of step
  }
}

// ---------------------------------------------------------------------------
// Kernel 3: decoder GEMM  out[s][t] = X[s][:] . W_dec[t][:] + b_dec[t]
// fp32 WMMA. Block tile 256x128 (16 waves, each wave 4x2 accumulators =
// 64x32), KT=32 K-slices staged in LDS. Only 8 row passes over W_dec (327MB
// > L2, so minimizing HBM re-reads dominates; X is L2-resident).
// ---------------------------------------------------------------------------
#define KT 32
__global__ __launch_bounds__(512) void k_decode_gemm(const float* __restrict__ X,
                                                     const float* __restrict__ Wd,
                                                     const float* __restrict__ bd,
                                                     float* __restrict__ out) {
  __shared__ float Xs[256][KT + 1];
  __shared__ float Ws[128][KT + 1];

  const int tid = threadIdx.x;
  const int lane = tid & 31;
  const int wave = tid >> 5;           // 0..15
  const int wr = wave >> 2;            // 0..3 : rows [wr*64, wr*64+64)
  const int wc = wave & 3;             // 0..3 : cols [wc*32, wc*32+32)
  const int row0 = blockIdx.y * 256;
  const int col0 = blockIdx.x * 128;

  v8f acc[4][2];
#pragma unroll
  for (int rt = 0; rt < 4; ++rt)
#pragma unroll
    for (int ct = 0; ct < 2; ++ct) acc[rt][ct] = (v8f){0.f,0.f,0.f,0.f,0.f,0.f,0.f,0.f};

  const int m  = lane & 15;            // f32 A/B operand layout (16x4):
  const int ks = (lane >> 4) << 1;     // lanes 0-15 K={0,1}, lanes 16-31 K={2,3}

  for (int kt = 0; kt < DEC_IN; kt += KT) {
    for (int t = tid; t < 256 * KT; t += 512) {
      int r = t >> 5, k = t & (KT - 1);
      Xs[r][k] = X[(size_t)(row0 + r) * DEC_IN + kt + k];
    }
    for (int t = tid; t < 128 * KT; t += 512) {
      int r = t >> 5, k = t & (KT - 1);
      const float* wp = &Wd[(size_t)(col0 + r) * DEC_IN + kt + k];
      Ws[r][k] = *wp;
      if (kt + KT < DEC_IN) __builtin_prefetch(wp + KT, 0, 1);  // global_prefetch_b8
    }
    __syncthreads();
#pragma unroll
    for (int k = 0; k < KT; k += 4) {
      v2f a[4], b[2];
#pragma unroll
      for (int rt = 0; rt < 4; ++rt) {
        a[rt].x = Xs[wr * 64 + rt * 16 + m][k + ks];
        a[rt].y = Xs[wr * 64 + rt * 16 + m][k + ks + 1];
      }
#pragma unroll
      for (int ct = 0; ct < 2; ++ct) {
        b[ct].x = Ws[wc * 32 + ct * 16 + m][k + ks];
        b[ct].y = Ws[wc * 32 + ct * 16 + m][k + ks + 1];
      }
#pragma unroll
      for (int rt = 0; rt < 4; ++rt)
#pragma unroll
        for (int ct = 0; ct < 2; ++ct)
          acc[rt][ct] = __builtin_amdgcn_wmma_f32_16x16x4_f32(
              false, a[rt], false, b[ct], (short)0, acc[rt][ct], false, false);
    }
    __syncthreads();
  }

  // C/D layout: VGPR v -> M = v + 8*(lane>=16), N = lane % 16
#pragma unroll
  for (int ct = 0; ct < 2; ++ct) {
    const int ncol = col0 + wc * 32 + ct * 16 + (lane & 15);
    const float bn = bd[ncol];
#pragma unroll
    for (int rt = 0; rt < 4; ++rt) {
      const int mrow = row0 + wr * 64 + rt * 16 + ((lane >> 4) << 3);
#pragma unroll
      for (int v = 0; v < 8; ++v) {
        out[(size_t)(mrow + v) * NTOK + ncol] = acc[rt][ct][v] + bn;
      }
    }
  }
}

// ---------------------------------------------------------------------------
// Kernel 4: in-place row-wise log_softmax over 32000 logits.
// ---------------------------------------------------------------------------
__global__ __launch_bounds__(256) void k_logsoftmax(float* __restrict__ out) {
  const int s = blockIdx.x;
  float* row = out + (size_t)s * NTOK;
  __shared__ float red[8];
  const int tid = threadIdx.x;
  const int lane = tid & 31;
  const int wave = tid >> 5;

  float m = -INFINITY;
  for (int t = tid; t < NTOK; t += 256) m = fmaxf(m, row[t]);
  m = wave_max32(m);
  if (lane == 0) red[wave] = m;
  __syncthreads();
  float gm = red[0];
  for (int i = 1; i < 8; ++i) gm = fmaxf(gm, red[i]);
  __syncthreads();

  float ssum = 0.0f;
  for (int t = tid; t < NTOK; t += 256) ssum += expf(row[t] - gm);
  ssum = wave_sum32(ssum);
  if (lane == 0) red[wave] = ssum;
  __syncthreads();
  float Z = 0.0f;
  for (int i = 0; i < 8; ++i) Z += red[i];
  const float lse = gm + logf(Z);

  for (int t = tid; t < NTOK; t += 256) row[t] -= lse;
}

// ---------------------------------------------------------------------------
extern "C" void kernel_launch(void* const* d_in, const int* in_sizes, int n_in,
                              void* d_out, int out_size, void* d_ws, size_t ws_size,
                              hipStream_t stream) {
  (void)in_sizes; (void)n_in; (void)out_size; (void)ws_size;
  const int*   ids  = (const int*)d_in[0];
  const float* encW = (const float*)d_in[1];
  const float* Wq1  = (const float*)d_in[2];
  const float* bq1  = (const float*)d_in[3];
  const float* Wq2  = (const float*)d_in[4];
  const float* bq2  = (const float*)d_in[5];
  const float* kbK  = (const float*)d_in[6];
  const float* kbV  = (const float*)d_in[7];
  const float* Wih  = (const float*)d_in[8];
  const float* bih  = (const float*)d_in[9];
  const float* Whh  = (const float*)d_in[10];
  const float* bhh  = (const float*)d_in[11];
  const float* Wd   = (const float*)d_in[12];
  const float* bd   = (const float*)d_in[13];
  float* out = (float*)d_out;

  // workspace layout
  float* X       = (float*)d_ws;                    // [SEQ][DEC_IN] ~21MB
  float* qh_part = X + (size_t)SEQ * DEC_IN;        // [8][QIN]
  float* q_part  = qh_part + 8 * QIN;               // [NBLK][QUERY]
  float* vpart_g = q_part + NBLK * QUERY;           // [NBLK][VALUE]
  float* Zpart_g = vpart_g + NBLK * VALUE;          // [NBLK]
  float* hx_g    = Zpart_g + NBLK;                  // [STATE]
  float* cx_g    = hx_g + STATE;                    // [STATE]
  unsigned* cnt  = (unsigned*)(cx_g + STATE + 64);  // barrier counter

  k_init<<<1, 256, 0, stream>>>(cnt, hx_g, cx_g);
  k_embed<<<SEQ, 256, 0, stream>>>(ids, encW, X);
  k_scan<<<NBLK, 1024, 0, stream>>>(Wq1, bq1, Wq2, bq2, kbK, kbV,
                                    Wih, bih, Whh, bhh, X,
                                    qh_part, q_part, vpart_g, Zpart_g,
                                    hx_g, cx_g, cnt);
  dim3 grid(NTOK / 128, SEQ / 256);
  k_decode_gemm<<<grid, 512, 0, stream>>>(X, Wd, bd, out);
  k_logsoftmax<<<SEQ, 256, 0, stream>>>(out);
}